// SessionGraph_26611617366069
// MI455X (gfx1250) — compile-verified
//
#include <hip/hip_runtime.h>
#include <hip/hip_bf16.h>
#include <stdint.h>

// ---------------- problem constants ----------------
constexpr int Hh    = 256;
constexpr int Bsz   = 1024;
constexpr int Ls    = 50;      // L
constexpr int LUs   = 50;      // LU
constexpr int NNn   = Bsz * LUs;     // 51200 graph nodes
constexpr int NE    = NNn * 12;      // 614400 edges
constexpr int RET_N = 10000;
constexpr int TOPKn = 10;
constexpr int MBITS = 64;
constexpr int NREL  = 4;

typedef __attribute__((ext_vector_type(2))) float v2f;
typedef __attribute__((ext_vector_type(8))) float v8f;

// ---------------- helpers ----------------
__device__ __forceinline__ unsigned f2o(float f) {
  unsigned u = __float_as_uint(f);
  return (u & 0x80000000u) ? ~u : (u | 0x80000000u);
}
__device__ __forceinline__ float o2f(unsigned u) {
  unsigned v = (u & 0x80000000u) ? (u & 0x7FFFFFFFu) : ~u;
  return __uint_as_float(v);
}

// ---------------- WMMA f32 GEMM: C[M,N] = A[M,K] * B + (bias) (+C if ACC) ----
// Each wave computes a 32x32 macro-tile (2 M-tiles x 2 N-tiles) so every
// A/B fragment load feeds two WMMAs: 4 loads -> 4 v_wmma per K-slice.
// WT_T = true : weight stored [N,K] row-major (x @ W.T)  -> B[k][n] = W[n*K+k]
// WT_T = false: weight stored [K,N] row-major (x @ W)    -> B[k][n] = W[k*N+n]
template <bool WT_T, bool BIAS, bool ACC>
__global__ void k_gemm(const float* __restrict__ A, const float* __restrict__ W,
                       const float* __restrict__ bias, float* __restrict__ C,
                       int M, int N, int K) {
  int wave = (blockIdx.x * blockDim.x + threadIdx.x) >> 5;
  int lane = threadIdx.x & 31;
  int n2  = N >> 5;                   // 32-wide N macro-tiles
  int mt2 = ((M >> 4) + 1) >> 1;      // 32-tall M macro-tiles (ceil over 16-tiles)
  if (wave >= mt2 * n2) return;
  int m0 = (wave / n2) << 5;
  int n0 = (wave % n2) << 5;
  bool hasM2 = (m0 + 16) < M;         // second M-tile valid? (M always mult of 16)
  int half = lane >> 4;               // selects K pair {k,k+1} vs {k+2,k+3}
  int l16  = lane & 15;

  v8f acc00 = {}, acc01 = {}, acc10 = {}, acc11 = {};
  const float* arow0 = A + (size_t)(m0 + l16) * K;
  const float* arow1 = A + (size_t)(m0 + (hasM2 ? 16 : 0) + l16) * K;

  for (int k = 0; k < K; k += 4) {
    int ka = k + 2 * half;
    v2f a0, a1, b0, b1;
    a0.x = arow0[ka]; a0.y = arow0[ka + 1];
    a1.x = arow1[ka]; a1.y = arow1[ka + 1];
    if (WT_T) {
      const float* w0 = W + (size_t)(n0 + l16) * K;
      const float* w1 = W + (size_t)(n0 + 16 + l16) * K;
      b0.x = w0[ka]; b0.y = w0[ka + 1];
      b1.x = w1[ka]; b1.y = w1[ka + 1];
    } else {
      b0.x = W[(size_t)ka * N + (n0 + l16)];
      b0.y = W[(size_t)(ka + 1) * N + (n0 + l16)];
      b1.x = W[(size_t)ka * N + (n0 + 16 + l16)];
      b1.y = W[(size_t)(ka + 1) * N + (n0 + 16 + l16)];
    }
    acc00 = __builtin_amdgcn_wmma_f32_16x16x4_f32(false, a0, false, b0, (short)0,
                                                  acc00, false, false);
    acc01 = __builtin_amdgcn_wmma_f32_16x16x4_f32(false, a0, false, b1, (short)0,
                                                  acc01, false, false);
    acc10 = __builtin_amdgcn_wmma_f32_16x16x4_f32(false, a1, false, b0, (short)0,
                                                  acc10, false, false);
    acc11 = __builtin_amdgcn_wmma_f32_16x16x4_f32(false, a1, false, b1, (short)0,
                                                  acc11, false, false);
  }

  float bb0 = BIAS ? bias[n0 + l16] : 0.0f;
  float bb1 = BIAS ? bias[n0 + 16 + l16] : 0.0f;
  #pragma unroll
  for (int v = 0; v < 8; ++v) {
    int row0 = m0 + v + 8 * half;
    size_t i00 = (size_t)row0 * N + (n0 + l16);
    size_t i01 = (size_t)row0 * N + (n0 + 16 + l16);
    float r00 = acc00[v] + bb0;
    float r01 = acc01[v] + bb1;
    if (ACC) { r00 += C[i00]; r01 += C[i01]; }
    C[i00] = r00;
    C[i01] = r01;
    if (hasM2) {
      int row1 = row0 + 16;
      size_t i10 = (size_t)row1 * N + (n0 + l16);
      size_t i11 = (size_t)row1 * N + (n0 + 16 + l16);
      float r10 = acc10[v] + bb0;
      float r11 = acc11[v] + bb1;
      if (ACC) { r10 += C[i10]; r11 += C[i11]; }
      C[i10] = r10;
      C[i11] = r11;
    }
  }
}
static inline int gemm_grid(int M, int N) {
  long long tiles = (long long)(((M / 16) + 1) / 2) * (N / 32);  // waves
  return (int)((tiles * 32 + 255) / 256);
}

// ---------------- elementwise / gather kernels ----------------
__global__ void k_means(const int* __restrict__ items, const int* __restrict__ cates,
                        const int* __restrict__ mask,
                        const float* __restrict__ item_emb, const float* __restrict__ cate_emb,
                        float* __restrict__ mean_i, float* __restrict__ mean_c,
                        int* __restrict__ lens) {
  int b = blockIdx.x, h = threadIdx.x;
  float ai = 0.f, ac = 0.f, cnt = 0.f;
  for (int j = 0; j < LUs; ++j) {
    int it = items[b * LUs + j];
    float g = (it != 0) ? 1.f : 0.f;
    ai += g * item_emb[(size_t)it * Hh + h];
    ac += g * cate_emb[(size_t)cates[b * LUs + j] * Hh + h];
    cnt += g;
  }
  mean_i[(size_t)b * Hh + h] = ai / cnt;
  mean_c[(size_t)b * Hh + h] = ac / cnt;
  if (h == 0) {
    int s = 0;
    for (int j = 0; j < Ls; ++j) s += mask[b * Ls + j];
    lens[b] = s;
  }
}

__global__ void k_seq(const int* __restrict__ alias_, const int* __restrict__ items,
                      const int* __restrict__ mask, const int* __restrict__ lens,
                      const float* __restrict__ item_emb, const float* __restrict__ pos_emb,
                      const float* __restrict__ len_emb, float* __restrict__ seq) {
  int bl = blockIdx.x, h = threadIdx.x;
  int b = bl / Ls, l = bl % Ls;
  int len = lens[b];
  int m = mask[b * Ls + l];
  int pos = (len - 1 - l) * m;
  int a = alias_[b * Ls + l];
  int it = items[b * LUs + a];
  seq[(size_t)bl * Hh + h] = item_emb[(size_t)it * Hh + h] +
                             pos_emb[(size_t)pos * Hh + h] +
                             len_emb[(size_t)len * Hh + h];
}

__global__ void k_ht(const float* __restrict__ seq, const int* __restrict__ lens,
                     float* __restrict__ ht) {
  int b = blockIdx.x, h = threadIdx.x;
  int last = lens[b] - 1;
  ht[(size_t)b * Hh + h] = seq[(size_t)(b * Ls + last) * Hh + h];
}

__global__ void k_alpha(const float* __restrict__ qsum, const float* __restrict__ q3,
                        const float* __restrict__ seq, const int* __restrict__ mask,
                        const float* __restrict__ five_w, const float* __restrict__ ht,
                        float* __restrict__ cat) {
  int b = blockIdx.x, t = threadIdx.x;
  __shared__ float al[Ls];
  if (t < Ls) {
    float s = 0.f;
    const float* q3r = q3 + (size_t)(b * Ls + t) * Hh;
    const float* qs = qsum + (size_t)b * Hh;
    for (int h = 0; h < Hh; ++h) {
      float pre = qs[h] + q3r[h];
      float sg = 1.f / (1.f + expf(-pre));
      s += sg * five_w[h];
    }
    al[t] = s;
  }
  __syncthreads();
  float acc = 0.f;
  for (int l = 0; l < Ls; ++l) {
    float m = (float)mask[b * Ls + l];
    acc += al[l] * seq[(size_t)(b * Ls + l) * Hh + t] * m;
  }
  cat[(size_t)b * 2 * Hh + t] = acc;
  cat[(size_t)b * 2 * Hh + Hh + t] = ht[(size_t)b * Hh + t];
}

__global__ void k_gather_x(const int* __restrict__ items, const float* __restrict__ item_emb,
                           float* __restrict__ x) {
  size_t i = (size_t)blockIdx.x * blockDim.x + threadIdx.x;
  if (i >= (size_t)NNn * Hh) return;
  int n = (int)(i >> 8), h = (int)(i & 255);
  x[i] = item_emb[(size_t)items[n] * Hh + h];
}

__global__ void k_sdots(const float* __restrict__ x, const float* __restrict__ a_src,
                        const float* __restrict__ a_dst, float* __restrict__ s_src,
                        float* __restrict__ s_dst) {
  int warp = threadIdx.x >> 5, lane = threadIdx.x & 31;
  int n = blockIdx.x * 8 + warp;
  const float* xr = x + (size_t)n * Hh;
  float ss = 0.f, sd = 0.f;
  for (int h = lane; h < Hh; h += 32) {
    float v = xr[h];
    ss += v * a_src[h];
    sd += v * a_dst[h];
  }
  for (int o = 16; o > 0; o >>= 1) {
    ss += __shfl_down(ss, o, 32);
    sd += __shfl_down(sd, o, 32);
  }
  if (lane == 0) { s_src[n] = ss; s_dst[n] = sd; }
}

__global__ void k_relprep(const float* __restrict__ rel_emb, const float* __restrict__ W,
                          const float* __restrict__ a_rel, float* __restrict__ rW,
                          float* __restrict__ srel) {
  int r = blockIdx.x, t = threadIdx.x;
  float acc = 0.f;
  for (int k = 0; k < Hh; ++k) acc += rel_emb[r * Hh + k] * W[(size_t)k * Hh + t];
  rW[r * Hh + t] = acc;
  __shared__ float red[256];
  red[t] = rel_emb[r * Hh + t] * a_rel[t];
  __syncthreads();
  for (int s = 128; s > 0; s >>= 1) {
    if (t < s) red[t] += red[t + s];
    __syncthreads();
  }
  if (t == 0) srel[r] = red[0];
}

__global__ void k_edge1(const int* __restrict__ ei, const int* __restrict__ et,
                        const float* __restrict__ s_src, const float* __restrict__ s_dst,
                        const float* __restrict__ srel, float* __restrict__ lw,
                        unsigned* __restrict__ emax) {
  int e = blockIdx.x * blockDim.x + threadIdx.x;
  if (e >= NE) return;
  int s = ei[e], d = ei[NE + e];
  float v = s_src[s] + s_dst[d] + srel[et[e]];
  v = (v > 0.f) ? v : 0.2f * v;   // leaky_relu(0.2)
  lw[e] = v;
  atomicMax(&emax[d], f2o(v));
}

__global__ void k_edge2(const int* __restrict__ ei, float* __restrict__ lw,
                        const unsigned* __restrict__ emax, float* __restrict__ denom) {
  int e = blockIdx.x * blockDim.x + threadIdx.x;
  if (e >= NE) return;
  int d = ei[NE + e];
  float w = expf(lw[e] - o2f(emax[d]));
  lw[e] = w;
  atomicAdd(&denom[d], w);
}

__global__ void k_edge3(const int* __restrict__ ei, const int* __restrict__ et,
                        const float* __restrict__ lw, const float* __restrict__ denom,
                        const float* __restrict__ y, const float* __restrict__ rW,
                        float* __restrict__ agg) {
  int e = blockIdx.x, h = threadIdx.x;
  int s = ei[e], d = ei[NE + e], r = et[e];
  float alpha = lw[e] / (denom[d] + 1e-16f);
  float val = alpha * (y[(size_t)s * Hh + h] + rW[r * Hh + h]);
  atomicAdd(&agg[(size_t)d * Hh + h], val);
}

__global__ void k_edge4(const float* __restrict__ agg, const float* __restrict__ xself,
                        float* __restrict__ gx) {
  size_t i = (size_t)blockIdx.x * blockDim.x + threadIdx.x;
  if (i >= (size_t)NNn * Hh) return;
  float v = agg[i] + xself[i];
  gx[i] = (v > 0.f) ? v : (expf(v) - 1.f);   // elu
}

__global__ void k_gseq(const int* __restrict__ alias_, const float* __restrict__ gx,
                       float* __restrict__ gseq) {
  int bl = blockIdx.x, h = threadIdx.x;
  int b = bl / Ls, l = bl % Ls;
  int a = alias_[b * Ls + l];
  gseq[(size_t)bl * Hh + h] = gx[(size_t)(b * LUs + a) * Hh + h];
}

__global__ void k_pack(const float* __restrict__ prod, unsigned long long* __restrict__ bits,
                       int rows) {
  int r = blockIdx.x * blockDim.x + threadIdx.x;
  if (r >= rows) return;
  unsigned long long v = 0ULL;
  for (int j = 0; j < MBITS; ++j)
    if (prod[(size_t)r * MBITS + j] > 0.f) v |= (1ULL << j);
  bits[r] = v;
}

__global__ void k_topk(const unsigned long long* __restrict__ htb,
                       const unsigned long long* __restrict__ hnb,
                       const int* __restrict__ labels, float* __restrict__ out_sim,
                       float* __restrict__ out_lab) {
  int b = blockIdx.x, t = threadIdx.x;
  __shared__ unsigned char hamv[RET_N];
  __shared__ int cnt[MBITS + 1];
  __shared__ int off[MBITS + 1];
  __shared__ int sel[TOPKn];
  if (t <= MBITS) cnt[t] = 0;
  __syncthreads();
  unsigned long long hb = htb[b];
  for (int r = t; r < RET_N; r += blockDim.x) {
    int hm = __popcll(hb ^ hnb[r]);
    hamv[r] = (unsigned char)hm;
    atomicAdd(&cnt[hm], 1);
  }
  __syncthreads();
  if (t == 0) {
    int run = 0;
    for (int v = 0; v <= MBITS; ++v) { off[v] = run; run += cnt[v]; }
    int found = 0;
    for (int r = 0; r < RET_N && found < TOPKn; ++r) {
      int v = hamv[r];
      int p = off[v]++;              // rank in (ham asc, idx asc) order
      if (p < TOPKn) { sel[p] = r; found++; }
    }
    float wv[TOPKn];
    float mx = -1e30f;
    for (int p = 0; p < TOPKn; ++p) {
      wv[p] = -(float)hamv[sel[p]] / (float)MBITS;
      if (wv[p] > mx) mx = wv[p];
    }
    float ssum = 0.f;
    for (int p = 0; p < TOPKn; ++p) { wv[p] = expf(wv[p] - mx); ssum += wv[p]; }
    for (int p = 0; p < TOPKn; ++p) {
      out_sim[b * TOPKn + p] = wv[p] / ssum;
      out_lab[b * TOPKn + p] = (float)labels[sel[p]];
    }
  }
}

__global__ void k_ln(const float* __restrict__ f, float* __restrict__ out) {
  int b = blockIdx.x, t = threadIdx.x;
  __shared__ float red[256];
  float v = f[(size_t)b * Hh + t];
  red[t] = v;
  __syncthreads();
  for (int s = 128; s > 0; s >>= 1) { if (t < s) red[t] += red[t + s]; __syncthreads(); }
  float mu = red[0] / Hh;
  __syncthreads();
  float d = v - mu;
  red[t] = d * d;
  __syncthreads();
  for (int s = 128; s > 0; s >>= 1) { if (t < s) red[t] += red[t + s]; __syncthreads(); }
  float var = red[0] / Hh;
  out[(size_t)b * Hh + t] = (v - mu) * rsqrtf(var + 1e-5f);
}

// ---------------- host launch ----------------
extern "C" void kernel_launch(void* const* d_in, const int* in_sizes, int n_in,
                              void* d_out, int out_size, void* d_ws, size_t ws_size,
                              hipStream_t stream) {
  (void)in_sizes; (void)n_in; (void)out_size; (void)ws_size;
  // inputs: recursive insertion-order flattening of setup_inputs()
  const int* alias_   = (const int*)d_in[0];
  const int* items    = (const int*)d_in[1];
  const int* mask     = (const int*)d_in[2];
  const int* cates    = (const int*)d_in[3];
  const int* ei       = (const int*)d_in[4];
  const int* etype    = (const int*)d_in[5];
  const float* item_emb = (const float*)d_in[6];
  const float* cate_emb = (const float*)d_in[7];
  const float* pos_emb  = (const float*)d_in[8];
  const float* len_emb  = (const float*)d_in[9];
  const float* rel_emb  = (const float*)d_in[10];
  // enc params: [one.w, one.b, two.w, two.b, three.w, three.b, four.w, four.b,
  //              five_w, transform.w, transform.b]
  const float* e1[11]; for (int i = 0; i < 11; ++i) e1[i] = (const float*)d_in[11 + i];
  const float* e2[11]; for (int i = 0; i < 11; ++i) e2[i] = (const float*)d_in[22 + i];
  const float* g_asrc  = (const float*)d_in[33];
  const float* g_adst  = (const float*)d_in[34];
  const float* g_arel  = (const float*)d_in[35];
  const float* g_w     = (const float*)d_in[36];
  const float* g_wself = (const float*)d_in[37];
  const float* sess_emb = (const float*)d_in[38];
  const int*   his_lab  = (const int*)d_in[39];
  const float* hashM    = (const float*)d_in[40];

  float* outf = (float*)d_out;
  const size_t O_SIM = (size_t)Bsz * Hh;            // 262144
  const size_t O_LAB = O_SIM + (size_t)Bsz * TOPKn; // 272384

  // workspace allocator
  char* ws = (char*)d_ws;
  size_t off = 0;
  auto alloc = [&](size_t bytes) -> char* {
    char* p = ws + off;
    off = (off + bytes + 255) & ~(size_t)255;
    return p;
  };
  const size_t BIG = (size_t)NNn * Hh * sizeof(float);  // 52.4 MB
  float* bufA = (float*)alloc(BIG);   // q3 (enc1) -> y -> q3 (enc2)
  float* bufB = (float*)alloc(BIG);   // seq_hidden -> agg -> g_seq
  float* bufC = (float*)alloc(BIG);   // x -> graph_x
  float* bufD = (float*)alloc(BIG);   // xself
  float* w_edge = (float*)alloc((size_t)NE * sizeof(float));
  float* mean_i = (float*)alloc((size_t)Bsz * Hh * 4);
  float* mean_c = (float*)alloc((size_t)Bsz * Hh * 4);
  float* htb_f  = (float*)alloc((size_t)Bsz * Hh * 4);   // ht
  float* qsum   = (float*)alloc((size_t)Bsz * Hh * 4);
  float* finalb = (float*)alloc((size_t)Bsz * Hh * 4);
  float* catb   = (float*)alloc((size_t)Bsz * 2 * Hh * 4);
  float* hsbuf  = (float*)alloc((size_t)RET_N * MBITS * 4);
  float* hfbuf  = (float*)alloc((size_t)Bsz * MBITS * 4);
  unsigned long long* hnbits = (unsigned long long*)alloc((size_t)RET_N * 8);
  unsigned long long* htbits = (unsigned long long*)alloc((size_t)Bsz * 8);
  int* lens     = (int*)alloc((size_t)Bsz * 4);
  float* s_src  = (float*)alloc((size_t)NNn * 4);
  float* s_dst  = (float*)alloc((size_t)NNn * 4);
  unsigned* emax = (unsigned*)alloc((size_t)NNn * 4);
  float* denom  = (float*)alloc((size_t)NNn * 4);
  float* rW     = (float*)alloc((size_t)NREL * Hh * 4);
  float* srel   = (float*)alloc(256);

  const int eGrid = (NE + 255) / 256;
  const int nhGrid = (int)(((size_t)NNn * Hh + 255) / 256);

  // ---- means / lengths / seq_hidden ----
  k_means<<<Bsz, 256, 0, stream>>>(items, cates, mask, item_emb, cate_emb,
                                   mean_i, mean_c, lens);
  k_seq<<<Bsz * Ls, 256, 0, stream>>>(alias_, items, mask, lens, item_emb,
                                      pos_emb, len_emb, bufB);
  k_ht<<<Bsz, 256, 0, stream>>>(bufB, lens, htb_f);

  // ---- encoder 1 ----
  k_gemm<true, true, false><<<gemm_grid(NNn, Hh), 256, 0, stream>>>(
      bufB, e1[4], e1[5], bufA, NNn, Hh, Hh);                       // q3
  k_gemm<true, true, false><<<gemm_grid(Bsz, Hh), 256, 0, stream>>>(
      htb_f, e1[0], e1[1], qsum, Bsz, Hh, Hh);                      // q1
  k_gemm<true, true, true><<<gemm_grid(Bsz, Hh), 256, 0, stream>>>(
      mean_i, e1[2], e1[3], qsum, Bsz, Hh, Hh);                     // +q2
  k_gemm<true, true, true><<<gemm_grid(Bsz, Hh), 256, 0, stream>>>(
      mean_c, e1[6], e1[7], qsum, Bsz, Hh, Hh);                     // +q4
  k_alpha<<<Bsz, 256, 0, stream>>>(qsum, bufA, bufB, mask, e1[8], htb_f, catb);
  k_gemm<true, true, false><<<gemm_grid(Bsz, Hh), 256, 0, stream>>>(
      catb, e1[9], e1[10], finalb, Bsz, Hh, 2 * Hh);                // local_sess

  // ---- GAT ----
  k_gather_x<<<nhGrid, 256, 0, stream>>>(items, item_emb, bufC);
  k_gemm<false, false, false><<<gemm_grid(NNn, Hh), 256, 0, stream>>>(
      bufC, g_w, nullptr, bufA, NNn, Hh, Hh);                       // y = x @ W
  k_gemm<false, false, false><<<gemm_grid(NNn, Hh), 256, 0, stream>>>(
      bufC, g_wself, nullptr, bufD, NNn, Hh, Hh);                   // x @ W_self
  k_sdots<<<NNn / 8, 256, 0, stream>>>(bufC, g_asrc, g_adst, s_src, s_dst);
  k_relprep<<<NREL, 256, 0, stream>>>(rel_emb, g_w, g_arel, rW, srel);
  hipMemsetAsync(emax, 0, (size_t)NNn * 4, stream);
  hipMemsetAsync(denom, 0, (size_t)NNn * 4, stream);
  hipMemsetAsync(bufB, 0, BIG, stream);                             // agg = 0
  k_edge1<<<eGrid, 256, 0, stream>>>(ei, etype, s_src, s_dst, srel, w_edge, emax);
  k_edge2<<<eGrid, 256, 0, stream>>>(ei, w_edge, emax, denom);
  k_edge3<<<NE, 256, 0, stream>>>(ei, etype, w_edge, denom, bufA, rW, bufB);
  k_edge4<<<nhGrid, 256, 0, stream>>>(bufB, bufD, bufC);            // graph_x
  k_gseq<<<Bsz * Ls, 256, 0, stream>>>(alias_, bufC, bufB);         // g_seq
  k_ht<<<Bsz, 256, 0, stream>>>(bufB, lens, htb_f);

  // ---- encoder 2 (accumulates into finalb) ----
  k_gemm<true, true, false><<<gemm_grid(NNn, Hh), 256, 0, stream>>>(
      bufB, e2[4], e2[5], bufA, NNn, Hh, Hh);
  k_gemm<true, true, false><<<gemm_grid(Bsz, Hh), 256, 0, stream>>>(
      htb_f, e2[0], e2[1], qsum, Bsz, Hh, Hh);
  k_gemm<true, true, true><<<gemm_grid(Bsz, Hh), 256, 0, stream>>>(
      mean_i, e2[2], e2[3], qsum, Bsz, Hh, Hh);
  k_gemm<true, true, true><<<gemm_grid(Bsz, Hh), 256, 0, stream>>>(
      mean_c, e2[6], e2[7], qsum, Bsz, Hh, Hh);
  k_alpha<<<Bsz, 256, 0, stream>>>(qsum, bufA, bufB, mask, e2[8], htb_f, catb);
  k_gemm<true, true, true><<<gemm_grid(Bsz, Hh), 256, 0, stream>>>(
      catb, e2[9], e2[10], finalb, Bsz, Hh, 2 * Hh);                // final_sess

  // ---- LSH retrieval ----
  k_gemm<false, false, false><<<gemm_grid(RET_N, MBITS), 256, 0, stream>>>(
      sess_emb, hashM, nullptr, hsbuf, RET_N, MBITS, Hh);
  k_gemm<false, false, false><<<gemm_grid(Bsz, MBITS), 256, 0, stream>>>(
      finalb, hashM, nullptr, hfbuf, Bsz, MBITS, Hh);
  k_pack<<<(RET_N + 255) / 256, 256, 0, stream>>>(hsbuf, hnbits, RET_N);
  k_pack<<<(Bsz + 255) / 256, 256, 0, stream>>>(hfbuf, htbits, Bsz);
  k_topk<<<Bsz, 256, 0, stream>>>(htbits, hnbits, his_lab, outf + O_SIM, outf + O_LAB);

  // ---- layer-norm output ----
  k_ln<<<Bsz, 256, 0, stream>>>(finalb, outf);
}